// VLGraph_37546604102432
// MI455X (gfx1250) — compile-verified
//
#include <hip/hip_runtime.h>
#include <hip/hip_bf16.h>
#include <math.h>

// ---------------------------------------------------------------------------
// RGAT (GCE-GNN style) forward for MI455X / gfx1250, wave32 + f32 WMMA.
//   B=256 batches, N=150 nodes (padded to NP=160), D=128, L=50, R=4, 2 layers.
//   All GEMM-shaped work routed through V_WMMA_F32_16X16X4_F32 (exact fp32).
//   Phase-2 score GEMM: one relation per wave, A-fragments cached in VGPRs.
// ---------------------------------------------------------------------------

typedef __attribute__((ext_vector_type(2))) float v2f;
typedef __attribute__((ext_vector_type(8))) float v8f;

#define NB    256      // batch
#define NN    150      // real node count
#define NP    160      // padded node count (10 tiles of 16)
#define DD    128      // hidden dim
#define LL    50       // seq len (pos matrix inner dim)
#define RR    4        // relation count
#define K3    384      // 3*DD concat width
#define NT    10       // node tiles per batch

// f32 WMMA fragment layouts (ISA 7.12.2):
// A 16x4 f32:  lanes 0-15 -> M=lane, vgpr0=K+0, vgpr1=K+1
//              lanes16-31 -> M=lane-16, vgpr0=K+2, vgpr1=K+3
// B 4x16 f32:  mirrored with N in place of M.
// C/D 16x16:   vgpr v: lanes0-15 -> (M=v, N=lane); lanes16-31 -> (M=v+8, N=lane-16)

__device__ __forceinline__ v8f wmma_f32(v2f a, v2f b, v8f c) {
  return __builtin_amdgcn_wmma_f32_16x16x4_f32(false, a, false, b,
                                               (short)0, c, false, false);
}

// ---------------------------------------------------------------------------
// Kernel 1: gather embeddings + soft-position embedding + 384->128 projection
// One block per (batch, 16-node tile); 8 waves each own a 16-col output tile.
// ---------------------------------------------------------------------------
__global__ __launch_bounds__(256) void feat_proj_kernel(
    const int*   __restrict__ nodes,   // [B,N]
    const int*   __restrict__ ntype,   // [B,N]
    const float* __restrict__ posmat,  // [B,N,L]
    const float* __restrict__ emb,     // [VOCAB,D]
    const float* __restrict__ post,    // [200,D] (use rows 0..L-1)
    const float* __restrict__ typet,   // [4,D]
    const float* __restrict__ W,       // [384,D]
    float*       __restrict__ h0)      // [B,NP,D]
{
  __shared__ float Xs[16][388];        // padded stride: 388 % 64 == 4
  __shared__ float possum[16];

  const int tid = threadIdx.x;
  const int b  = blockIdx.x / NT;
  const int n0 = (blockIdx.x % NT) * 16;

  if (tid < 16) {
    const int n = n0 + tid;
    float s = 0.f;
    if (n < NN) {
      const float* p = posmat + ((size_t)b * NN + n) * LL;
      #pragma unroll 5
      for (int l = 0; l < LL; ++l) s += p[l];
    }
    possum[tid] = s + 1e-9f;
  }
  __syncthreads();

  // build X = [emb | type | pos] rows in LDS
  for (int idx = tid; idx < 16 * K3; idx += 256) {
    const int m = idx / K3, c = idx % K3;
    const int n = n0 + m;
    float v = 0.f;
    if (n < NN) {
      if (c < DD) {
        v = emb[(size_t)nodes[b * NN + n] * DD + c];
      } else if (c < 2 * DD) {
        v = typet[(size_t)ntype[b * NN + n] * DD + (c - DD)];
      } else {
        const int d = c - 2 * DD;
        const float* p = posmat + ((size_t)b * NN + n) * LL;
        float acc = 0.f;
        #pragma unroll 5
        for (int l = 0; l < LL; ++l) acc += p[l] * post[l * DD + d];
        const float valid = (ntype[b * NN + n] > 0) ? 1.f : 0.f;
        v = (acc / possum[m]) * valid;
      }
    }
    Xs[m][c] = v;
  }
  __syncthreads();

  const int wave = tid >> 5, lane = tid & 31;
  const int half = lane >> 4, l15 = lane & 15;
  const int col0 = wave * 16;

  v8f acc = {};
  for (int kk = 0; kk < K3; kk += 4) {
    const int ka = kk + 2 * half;
    v2f a, bb;
    a.x  = Xs[l15][ka];
    a.y  = Xs[l15][ka + 1];
    bb.x = W[(size_t)ka * DD + col0 + l15];
    bb.y = W[(size_t)(ka + 1) * DD + col0 + l15];
    acc = wmma_f32(a, bb, acc);
  }

  float* outp = h0 + ((size_t)b * NP + n0) * DD;
  #pragma unroll
  for (int vv = 0; vv < 8; ++vv) {
    const int row = vv + 8 * half;
    outp[(size_t)row * DD + col0 + l15] = acc[vv];
  }
}

// ---------------------------------------------------------------------------
// Kernel 2: one fused RGAT layer.
//   scores (WMMA, fixed relation per wave, A-frags cached in VGPRs)
//   -> LeakyReLU -> gather by adj -> masked softmax
//   -> aggregation alpha @ H (WMMA) -> valid mask.
// One block per (batch, 16-row i-tile). LDS kept < 64KB.
// ---------------------------------------------------------------------------
__global__ __launch_bounds__(256) void rgat_layer_kernel(
    const int*   __restrict__ adj,     // [B,N,N]
    const int*   __restrict__ ntype,   // [B,N]
    const float* __restrict__ arel,    // [R,D]
    const float* __restrict__ hin,     // [B,NP,D]
    float*       __restrict__ hout)    // [B,NP,D]
{
  __shared__ float Hi[16][132];         // i-tile rows of h (132 % 64 == 4)
  __shared__ float Ss[RR][16][NP];      // per-relation leakyrelu'd logits
  __shared__ float Al[16][164];         // softmax alpha (164 % 64 == 36)
  __shared__ float red[16][16];
  __shared__ float rowmax[16], rowsum[16], validv[16];

  const int tid = threadIdx.x;
  const int b  = blockIdx.x / NT;
  const int i0 = (blockIdx.x % NT) * 16;
  const float* hb = hin + (size_t)b * NP * DD;

  for (int idx = tid; idx < 16 * DD; idx += 256) {
    const int m = idx >> 7, k = idx & 127;
    Hi[m][k] = hb[(size_t)(i0 + m) * DD + k];
  }
  if (tid < 16) {
    const int i = i0 + tid;
    validv[tid] = (i < NN && ntype[b * NN + i] > 0) ? 1.f : 0.f;
  }
  __syncthreads();

  const int wave = tid >> 5, lane = tid & 31;
  const int half = lane >> 4, l15 = lane & 15;

  // ---- phase 2: S_r[i-tile, all j] via WMMA ------------------------------
  // wave -> fixed relation r = wave&3; j-tile half = wave>>2 (5 tiles each).
  {
    const int r  = wave & 3;
    const int jh = wave >> 2;

    // Cache all 32 A-fragments (h_i (.) a_r) for this relation in VGPRs.
    v2f afrag[DD / 4];
    #pragma unroll
    for (int kk = 0; kk < DD; kk += 4) {
      const int ka = kk + 2 * half;
      v2f a;
      a.x = Hi[l15][ka]     * arel[r * DD + ka];
      a.y = Hi[l15][ka + 1] * arel[r * DD + ka + 1];
      afrag[kk >> 2] = a;
    }

    for (int q = 0; q < 5; ++q) {
      const int j0 = (jh * 5 + q) * 16;
      const float* bp = hb + (size_t)(j0 + l15) * DD + 2 * half;
      if (q < 4) {
        // pull next j-tile's B rows toward the WGP (global_prefetch_b8)
        __builtin_prefetch(hb + (size_t)(j0 + 16 + l15) * DD + 2 * half, 0, 1);
      }
      v8f acc = {};
      #pragma unroll
      for (int kk = 0; kk < DD; kk += 4) {
        v2f bb;
        bb.x = bp[kk];
        bb.y = bp[kk + 1];
        acc = wmma_f32(afrag[kk >> 2], bb, acc);
      }
      #pragma unroll
      for (int vv = 0; vv < 8; ++vv) {
        const int row = vv + 8 * half;
        float e = acc[vv];
        e = e > 0.f ? e : 0.2f * e;          // LeakyReLU(alpha=0.2)
        Ss[r][row][j0 + l15] = e;
      }
    }
  }
  __syncthreads();

  // ---- phase 3: gather-by-relation, mask, softmax over j ------------------
  {
    const int g = tid >> 4, l = tid & 15;  // group g handles row i0+g
    const int i = i0 + g;
    float sc[10];
    #pragma unroll
    for (int s = 0; s < 10; ++s) {
      const int j = l + 16 * s;
      float v = -1e9f;
      if (i < NN && j < NN) {
        const int a = adj[((size_t)b * NN + i) * NN + j];
        if (a > 0) v = Ss[a - 1][g][j];
      }
      sc[s] = v;
    }
    float lmax = sc[0];
    #pragma unroll
    for (int s = 1; s < 10; ++s) lmax = fmaxf(lmax, sc[s]);
    red[g][l] = lmax;
    __syncthreads();
    if (l == 0) {
      float m = red[g][0];
      for (int t = 1; t < 16; ++t) m = fmaxf(m, red[g][t]);
      rowmax[g] = m;
    }
    __syncthreads();
    const float m = rowmax[g];
    float ex[10], lsum = 0.f;
    #pragma unroll
    for (int s = 0; s < 10; ++s) {
      const int j = l + 16 * s;
      const float e = (i < NN && j < NN) ? __expf(sc[s] - m) : 0.f;
      ex[s] = e; lsum += e;
    }
    red[g][l] = lsum;
    __syncthreads();
    if (l == 0) {
      float s2 = 0.f;
      for (int t = 0; t < 16; ++t) s2 += red[g][t];
      rowsum[g] = s2 + 1e-30f;
    }
    __syncthreads();
    const float inv = 1.f / rowsum[g];
    #pragma unroll
    for (int s = 0; s < 10; ++s) Al[g][l + 16 * s] = ex[s] * inv;
  }
  __syncthreads();

  // ---- phase 4: h_new = alpha[16x160] @ H[160x128] via WMMA ---------------
  {
    const int col0 = wave * 16;
    v8f acc = {};
    for (int kk = 0; kk < NP; kk += 4) {
      const int ka = kk + 2 * half;
      v2f a, bb;
      a.x  = Al[l15][ka];
      a.y  = Al[l15][ka + 1];
      bb.x = hb[(size_t)ka * DD + col0 + l15];
      bb.y = hb[(size_t)(ka + 1) * DD + col0 + l15];
      acc = wmma_f32(a, bb, acc);
    }
    float* op = hout + ((size_t)b * NP + i0) * DD;
    #pragma unroll
    for (int vv = 0; vv < 8; ++vv) {
      const int row = vv + 8 * half;
      op[(size_t)row * DD + col0 + l15] = acc[vv] * validv[row];
    }
  }
}

// ---------------------------------------------------------------------------
// Kernel 3: strip node padding into d_out [B,150,128]
// ---------------------------------------------------------------------------
__global__ __launch_bounds__(256) void copy_out_kernel(
    const float* __restrict__ hfin, float* __restrict__ out)
{
  const size_t idx = (size_t)blockIdx.x * 256 + threadIdx.x;
  const size_t total = (size_t)NB * NN * DD;
  if (idx >= total) return;
  const int d = (int)(idx % DD);
  const size_t bn = idx / DD;
  const int n = (int)(bn % NN);
  const int b = (int)(bn / NN);
  out[idx] = hfin[((size_t)b * NP + n) * DD + d];
}

// ---------------------------------------------------------------------------
extern "C" void kernel_launch(void* const* d_in, const int* in_sizes, int n_in,
                              void* d_out, int out_size, void* d_ws, size_t ws_size,
                              hipStream_t stream) {
  (void)in_sizes; (void)n_in; (void)out_size; (void)ws_size;
  const int*   adj    = (const int*)  d_in[0];
  const int*   nodes  = (const int*)  d_in[1];
  const int*   ntype  = (const int*)  d_in[2];
  const float* posmat = (const float*)d_in[3];
  const float* emb    = (const float*)d_in[4];
  const float* post   = (const float*)d_in[5];
  const float* typet  = (const float*)d_in[6];
  const float* W      = (const float*)d_in[7];
  const float* arel   = (const float*)d_in[8];
  float*       out    = (float*)d_out;

  float* h0 = (float*)d_ws;                   // [B,NP,D]
  float* h1 = h0 + (size_t)NB * NP * DD;      // [B,NP,D]

  const dim3 blk(256);
  const dim3 grd(NB * NT);

  feat_proj_kernel<<<grd, blk, 0, stream>>>(nodes, ntype, posmat, emb, post,
                                            typet, W, h0);
  rgat_layer_kernel<<<grd, blk, 0, stream>>>(adj, ntype, arel, h0, h1);
  rgat_layer_kernel<<<grd, blk, 0, stream>>>(adj, ntype, arel, h1, h0);

  const size_t total = (size_t)NB * NN * DD;
  copy_out_kernel<<<(unsigned)((total + 255) / 256), blk, 0, stream>>>(h0, out);
}